// Preprocess_51024211476487
// MI455X (gfx1250) — compile-verified
//
#include <hip/hip_runtime.h>
#include <cstdint>

// ---------------------------------------------------------------------------
// Preprocessing pipeline for MI455X (gfx1250, wave32).
// Memory-bound gather/compact/segment-mean. CDNA5 paths used:
//   - TENSOR_LOAD_TO_LDS (TDM) for strided hand-region tile extraction
//   - GLOBAL_LOAD_ASYNC_TO_LDS_B64 for the divergent lips gather
//   - s_wait_tensorcnt / s_wait_asynccnt
//   - V_WMMA_F32_16X16X4_F32 ones-matrix column reduction in the segment mean
//   - wave32 shuffle reductions, DS atomics, global prefetch
// ---------------------------------------------------------------------------

#define T_FR    32768
#define N_LM    543
#define FSTR    (N_LM * 3)     // 1629 floats per frame
#define NSEG    512
#define NHAND   42             // 21 landmarks * 2 coords
#define NLIPCOL 40
#define NLIP    80             // 40 * 2
#define NFEAT   (NHAND + NLIP) // 122
#define CHUNK   4              // frames per wave in K1
#define K3ROWS  32             // rows staged per chunk in K3

typedef __attribute__((ext_vector_type(4))) unsigned int v4u;
typedef __attribute__((ext_vector_type(8))) int          v8i;
typedef __attribute__((ext_vector_type(4))) int          v4i;
typedef __attribute__((ext_vector_type(2))) float        v2f;
typedef __attribute__((ext_vector_type(8))) float        v8f;

// -------- CDNA5 async global->LDS (per-lane, ASYNCcnt path) ----------------
__device__ __forceinline__ void async_b64(uint32_t lds_off, const void* src) {
  asm volatile("global_load_async_to_lds_b64 %0, %1, off"
               :: "v"(lds_off), "v"((unsigned long long)(uintptr_t)src)
               : "memory");
}
__device__ __forceinline__ void wait_async0() {
  asm volatile("s_wait_asynccnt 0" ::: "memory");
}

// -------- CDNA5 Tensor Data Mover: 2D tile (dim0 elems x rows) -> LDS ------
// D# packing per CDNA5 ISA (group0: count/lds/global/type,
// group1: data_size=4B, tensor_dim0/1, tile_dim0/1, tensor_dim0_stride).
// This toolchain exposes the 6-arg builtin: (v4u, v8i, v4i, v4i, v8i, i32).
__device__ __forceinline__ void tdm_load_2d(uint32_t lds_addr, const void* gaddr,
                                            uint32_t dim0, uint32_t rows,
                                            uint32_t stride0) {
  const uint64_t ga = (uint64_t)(uintptr_t)gaddr;
  v4u g0;
  g0.x = 0x1u;                                        // count=1 (valid), user mode
  g0.y = lds_addr;                                    // lds_addr (bytes)
  g0.z = (uint32_t)(ga & 0xFFFFFFFFu);                // global_addr[31:0]
  g0.w = (uint32_t)((ga >> 32) & 0x1FFFFFFu) | (2u << 30); // global_addr[56:32] | type=2
  v8i g1;
  g1[0] = (int)(2u << 16);                            // data_size=2 (4 bytes)
  g1[1] = (int)((dim0 & 0xFFFFu) << 16);              // tensor_dim0[15:0]
  g1[2] = (int)(((dim0 >> 16) & 0xFFFFu) | ((rows & 0xFFFFu) << 16)); // dim0 hi | dim1 lo
  g1[3] = (int)(((rows >> 16) & 0xFFFFu) | ((dim0 & 0xFFFFu) << 16)); // dim1 hi | tile_dim0
  g1[4] = (int)(rows & 0xFFFFu);                      // tile_dim1 | tile_dim2=0
  g1[5] = (int)stride0;                               // tensor_dim0_stride[31:0]
  g1[6] = 0;                                          // stride0 hi | stride1 lo
  g1[7] = 0;
  v4i g2 = {0, 0, 0, 0};                              // 2D tensor: groups 2/3 unused
  v4i g3 = {0, 0, 0, 0};
  v8i g4 = {0, 0, 0, 0, 0, 0, 0, 0};                  // extra group (unused, zero)
  __builtin_amdgcn_tensor_load_to_lds(g0, g1, g2, g3, g4, 0);
}

// ---------------- K0: zero global accumulators -----------------------------
__global__ void k0_zero(float* colsum, int* colcnt) {
  int i = threadIdx.x;
  if (i < NLIP) { colsum[i] = 0.0f; colcnt[i] = 0; }
}

// ---------------- K1: per-frame hand transform + keep + lips gather --------
// One wave per CHUNK=4 frames. TDM pulls the two strided hand tiles into LDS;
// async per-lane loads pull the 40 scattered lips (x,y) pairs into LDS.
__global__ __launch_bounds__(256) void k1_frames(
    const float* __restrict__ frames, const int* __restrict__ lips_idx,
    float* __restrict__ hand, float* __restrict__ lips, int* __restrict__ keep,
    float* __restrict__ colsum, int* __restrict__ colcnt) {
  // per wave: lh 4x63 floats | rh 4x63 floats | lips 4x80 floats = 824 -> pad 832
  __shared__ float stage[8][832];
  __shared__ float lsum[NLIP];
  __shared__ int   lcnt[NLIP];
  __shared__ int   sidx[NLIPCOL];

  const int tid = threadIdx.x;
  if (tid < NLIP)    { lsum[tid] = 0.0f; lcnt[tid] = 0; }
  if (tid < NLIPCOL) sidx[tid] = lips_idx[tid];
  __syncthreads();

  const int wave = tid >> 5;
  const int lane = tid & 31;
  const int gw = blockIdx.x * 8 + wave;       // 1024 blocks * 8 waves = 8192 waves
  const int t0 = gw * CHUNK;                  // 8192 * 4 = 32768 frames, exact cover

  const uint32_t lhb = (uint32_t)(uintptr_t)&stage[wave][0];
  const uint32_t rhb = lhb + 252u * 4u;       // after 4x63 lh floats
  const uint32_t lpb = lhb + 504u * 4u;       // after 4x63 rh floats

  const float* fb0 = frames + (size_t)t0 * FSTR;
  __builtin_prefetch((const char*)fb0 + 468 * 12, 0, 0);

  // TDM: lh tile (landmarks 468..488 -> float offset 1404) and rh tile (522.. -> 1566),
  // 63 floats wide, 4 frame-rows, row stride 1629 floats.
  tdm_load_2d(lhb, (const char*)fb0 + 1404 * 4, 63u, (uint32_t)CHUNK, (uint32_t)FSTR);
  tdm_load_2d(rhb, (const char*)fb0 + 1566 * 4, 63u, (uint32_t)CHUNK, (uint32_t)FSTR);

  // Async per-lane lips gather overlaps with the TDM transfers.
  for (int sub = 0; sub < CHUNK; ++sub) {
    const char* fb = (const char*)(fb0 + sub * FSTR);
    const int c0 = lane;                        // lanes 0..31 -> cols 0..31
    async_b64(lpb + (uint32_t)(sub * 320 + c0 * 8), fb + sidx[c0] * 12);
    const int c1 = lane + 32;                   // lanes 0..7 -> cols 32..39
    if (c1 < NLIPCOL)
      async_b64(lpb + (uint32_t)(sub * 320 + c1 * 8), fb + sidx[c1] * 12);
  }

  __builtin_amdgcn_s_wait_tensorcnt(0);
  wait_async0();

  for (int sub = 0; sub < CHUNK; ++sub) {
    const int t = t0 + sub;
    float hx = 0.f, hy = 0.f;
    if (lane < 21) {
      const float* L = &stage[wave][sub * 63];
      const float* R = &stage[wave][252 + sub * 63];
      const float lhx = L[3 * lane], lhy = L[3 * lane + 1];
      const float rhx = R[3 * lane], rhy = R[3 * lane + 1];
      hx = ((lhx != lhx) ? 0.f : lhx)          + ((rhx != rhx) ? 0.f : (1.0f - rhx));
      hy = ((lhy != lhy) ? 0.f : (1.0f - lhy)) + ((rhy != rhy) ? 0.f : (1.0f - rhy));
      hand[(size_t)t * NHAND + 2 * lane]     = hx;
      hand[(size_t)t * NHAND + 2 * lane + 1] = hy;
    }
    // wave32 reduction; all summands >= 0, so (sum != 0) is order-independent.
    float s = hx + hy;
    for (int o = 16; o; o >>= 1) s += __shfl_xor(s, o, 32);
    const bool kp = (s != 0.0f);
    if (lane == 0) keep[t] = kp ? 1 : 0;

    const float* P = &stage[wave][504 + sub * 80];
    {
      const int c = lane;
      const float lx = P[2 * c], ly = P[2 * c + 1];
      if (c < NLIPCOL) {
        lips[(size_t)t * NLIP + 2 * c]     = lx;
        lips[(size_t)t * NLIP + 2 * c + 1] = ly;
        if (kp) {
          if (lx == lx) { atomicAdd(&lsum[2 * c],     lx); atomicAdd(&lcnt[2 * c],     1); }
          if (ly == ly) { atomicAdd(&lsum[2 * c + 1], ly); atomicAdd(&lcnt[2 * c + 1], 1); }
        }
      }
    }
    {
      const int c = lane + 32;
      if (c < NLIPCOL) {
        const float lx = P[2 * c], ly = P[2 * c + 1];
        lips[(size_t)t * NLIP + 2 * c]     = lx;
        lips[(size_t)t * NLIP + 2 * c + 1] = ly;
        if (kp) {
          if (lx == lx) { atomicAdd(&lsum[2 * c],     lx); atomicAdd(&lcnt[2 * c],     1); }
          if (ly == ly) { atomicAdd(&lsum[2 * c + 1], ly); atomicAdd(&lcnt[2 * c + 1], 1); }
        }
      }
    }
  }
  __syncthreads();
  if (tid < NLIP) {
    atomicAdd(&colsum[tid], lsum[tid]);
    atomicAdd(&colcnt[tid], lcnt[tid]);
  }
}

// ---------------- K2: prefix scan of keep -> frame index map, S, col means -
__global__ __launch_bounds__(1024) void k2_scan(
    const int* __restrict__ keep, int* __restrict__ fidx, int* __restrict__ Sout,
    const float* __restrict__ colsum, const int* __restrict__ colcnt,
    float* __restrict__ colmean) {
  __shared__ int ls[1024];
  const int tid = threadIdx.x;
  const int CH = T_FR / 1024;   // 32 frames per thread
  const int base = tid * CH;
  int local = 0;
  for (int i = 0; i < CH; ++i) local += keep[base + i];
  ls[tid] = local;
  __syncthreads();
  for (int off = 1; off < 1024; off <<= 1) {
    int v = ls[tid];
    int add = (tid >= off) ? ls[tid - off] : 0;
    __syncthreads();
    ls[tid] = v + add;
    __syncthreads();
  }
  int run = (tid == 0) ? 0 : ls[tid - 1];
  for (int i = 0; i < CH; ++i) {
    int t = base + i;
    if (keep[t]) { fidx[run] = t; ++run; }
  }
  if (tid == 1023) Sout[0] = ls[1023];
  if (tid < NLIP) {
    int c = colcnt[tid];
    colmean[tid] = (c == 0) ? 0.0f : colsum[tid] / (float)c;
  }
}

// ---------------- K3: per-segment mean via WMMA ones-reduction -------------
// seg(i) = max{k : floor(k*(S-1)/512) <= i}  =>  segment k owns i in [b_k, b_{k+1})
// with b_k = floor(k*(S-1)/512). Frame i = S-1 maps to segment 512 and is
// dropped by jax segment_sum -- our intervals exclude it naturally.
// Column sums computed with V_WMMA_F32_16X16X4_F32, A = ones(16x4):
// D[m][n] = sum_k B[k][n] (+C), identical for every m; invariant to the
// K<->(VGPR,lane-half) permutation, so only the N=lane&15 mapping matters.
__global__ __launch_bounds__(128) void k3_segmean(
    const int* __restrict__ fidx, const float* __restrict__ hand,
    const float* __restrict__ lips, const float* __restrict__ colmean,
    const int* __restrict__ Sp, float* __restrict__ mean) {
  __shared__ float tile[K3ROWS][144];   // 128 cols padded to 144 (bank-split halves)
  __shared__ int   sid[K3ROWS];

  const int k    = blockIdx.x;
  const int tid  = threadIdx.x;
  const int wave = tid >> 5;
  const int lane = tid & 31;
  const int half = lane >> 4;           // 0: rows r4,r4+1  1: rows r4+2,r4+3
  const int l15  = lane & 15;
  const int S = Sp[0];
  uint64_t b0 = 0, b1 = 0;
  if (S >= 2) {
    b0 = ((uint64_t)k       * (uint64_t)(S - 1)) >> 9;
    b1 = ((uint64_t)(k + 1) * (uint64_t)(S - 1)) >> 9;
  }
  const int cnt = (int)(b1 - b0);

  v2f ones; ones.x = 1.0f; ones.y = 1.0f;
  v8f accA = {0.f,0.f,0.f,0.f,0.f,0.f,0.f,0.f};
  v8f accB = accA;
  const int tA = wave * 2;              // feature tiles: wave handles tA, tA+1
  const int tB = wave * 2 + 1;

  for (int base = 0; base < cnt; base += K3ROWS) {
    if (tid < K3ROWS)
      sid[tid] = (base + tid < cnt) ? fidx[b0 + (uint64_t)(base + tid)] : -1;
    __syncthreads();
    // Stage K3ROWS x 128 raw values (NaN->colmean, zero pad rows/cols).
    for (int r = 0; r < K3ROWS; ++r) {
      float v = 0.f;
      const int t = sid[r];
      if (t >= 0 && tid < NFEAT) {
        if (tid < NHAND) {
          v = hand[(size_t)t * NHAND + tid];
        } else {
          v = lips[(size_t)t * NLIP + (tid - NHAND)];
          if (v != v) v = colmean[tid - NHAND];
        }
      }
      tile[r][tid] = v;
    }
    __syncthreads();
    // 8 WMMA k-steps of 4 rows each; EXEC is all-ones here (no divergence).
    for (int r4 = 0; r4 < K3ROWS; r4 += 4) {
      v2f bA, bB;
      bA.x = tile[r4 + 2 * half][tA * 16 + l15];
      bA.y = tile[r4 + 2 * half + 1][tA * 16 + l15];
      bB.x = tile[r4 + 2 * half][tB * 16 + l15];
      bB.y = tile[r4 + 2 * half + 1][tB * 16 + l15];
      accA = __builtin_amdgcn_wmma_f32_16x16x4_f32(
          false, ones, false, bA, (short)0, accA, false, false);
      accB = __builtin_amdgcn_wmma_f32_16x16x4_f32(
          false, ones, false, bB, (short)0, accB, false, false);
    }
  }

  // Every lane's acc[0] holds the column sum for N = lane&15 of its tile.
  if (half == 0) {
    int f = tA * 16 + l15;
    if (f < NFEAT)
      mean[(size_t)k * NFEAT + f] = (cnt == 0) ? 0.f : accA[0] / (float)cnt;
    f = tB * 16 + l15;
    if (f < NFEAT)
      mean[(size_t)k * NFEAT + f] = (cnt == 0) ? 0.f : accB[0] / (float)cnt;
  }
}

// ---------------- K4: drop all-zero rows, compact into d_out ---------------
__global__ __launch_bounds__(512) void k4_rows(const float* __restrict__ mean,
                                               float* __restrict__ out) {
  __shared__ int fl[NSEG];
  const int r = threadIdx.x;
  float s = 0.f;
  for (int c = 0; c < NFEAT; ++c) s += mean[(size_t)r * NFEAT + c];
  const int f = (s != 0.0f) ? 1 : 0;
  fl[r] = f;
  __syncthreads();
  for (int off = 1; off < NSEG; off <<= 1) {
    int v = fl[r];
    int add = (r >= off) ? fl[r - off] : 0;
    __syncthreads();
    fl[r] = v + add;
    __syncthreads();
  }
  if (f) {
    const int pos = fl[r] - 1;   // exclusive prefix
    for (int c = 0; c < NFEAT; ++c)
      out[(size_t)pos * NFEAT + c] = mean[(size_t)r * NFEAT + c];
  }
}

// ---------------------------------------------------------------------------
extern "C" void kernel_launch(void* const* d_in, const int* in_sizes, int n_in,
                              void* d_out, int out_size, void* d_ws, size_t ws_size,
                              hipStream_t stream) {
  const float* frames   = (const float*)d_in[0];   // (32768, 543, 3) f32
  const int*   lips_idx = (const int*)d_in[1];     // (40,) i32
  float*       out      = (float*)d_out;

  // Workspace layout (~16.5 MB)
  float* ws_f   = (float*)d_ws;
  float* hand   = ws_f;                                 // T_FR*42
  float* lips   = hand + (size_t)T_FR * NHAND;          // T_FR*80
  float* colsum = lips + (size_t)T_FR * NLIP;           // 80
  float* colmean= colsum + NLIP;                        // 80
  float* mean   = colmean + NLIP;                       // 512*122
  int*   keep   = (int*)(mean + (size_t)NSEG * NFEAT);  // T_FR
  int*   fidx   = keep + T_FR;                          // T_FR
  int*   colcnt = fidx + T_FR;                          // 80
  int*   Sp     = colcnt + NLIP;                        // 1

  k0_zero   <<<1,    256, 0, stream>>>(colsum, colcnt);
  k1_frames <<<1024, 256, 0, stream>>>(frames, lips_idx, hand, lips, keep, colsum, colcnt);
  k2_scan   <<<1,   1024, 0, stream>>>(keep, fidx, Sp, colsum, colcnt, colmean);
  k3_segmean<<<NSEG, 128, 0, stream>>>(fidx, hand, lips, colmean, Sp, mean);
  k4_rows   <<<1,    512, 0, stream>>>(mean, out);
  (void)in_sizes; (void)n_in; (void)out_size; (void)ws_size;
}